// GraphSage_5686536700270
// MI455X (gfx1250) — compile-verified
//
#include <hip/hip_runtime.h>

// ---------------------------------------------------------------------------
// GraphSAGE forward on MI455X (gfx1250, wave32).
//
// Roofline: GEMMs total ~4.2 GFLOP (negligible vs WMMA ceiling); the workload
// is dominated by the two edge-scatters (640K edges x 512B gathers + ~82M f32
// atomics each), whose 20.5MB working set is L2-resident (192MB L2). HBM
// traffic is only first-touch + streamed outputs (~100MB -> ~5us @ 23.3TB/s).
// Since math is free on this roofline we keep bit-exact fp32 and use
// V_WMMA_F32_16X16X4_F32 for all dense parts.
//
// vs round 1: k-loops made wave-uniform (koff folded into base pointers) so
// no per-iteration EXEC masking around WMMA, and K/M are template constants
// so the fully-unrolled loops use immediate-offset global loads (no per-iter
// 64-bit address VALU), letting loads be hoisted ahead of the serialized
// WMMA accumulation chain.
// ---------------------------------------------------------------------------

typedef __attribute__((ext_vector_type(2))) float v2f;
typedef __attribute__((ext_vector_type(8))) float v8f;

#define N_NODES 40000
#define FEAT    128     // aggregation width for both layers

// ---------------------------------------------------------------- zero fill
__global__ void gs_zero_kernel(float* __restrict__ p, int n) {
    int i = blockIdx.x * blockDim.x + threadIdx.x;
    if (i < n) p[i] = 0.0f;
}

// ------------------------------------------------------------ edge scatter
// One wave32 per edge: lane l handles 4 consecutive floats (float4 gather of
// x[src], 4 global f32 atomics into agg[dst]). Feature dim fixed at 128.
__global__ void gs_scatter_kernel(const float* __restrict__ x,
                                  const int*   __restrict__ src,
                                  const int*   __restrict__ dst,
                                  float* __restrict__ agg,
                                  float* __restrict__ deg,
                                  int n_edges, int add_deg) {
    int gid  = blockIdx.x * blockDim.x + threadIdx.x;
    int edge = gid >> 5;
    int lane = gid & 31;
    if (edge >= n_edges) return;
    int s = src[edge];
    int d = dst[edge];
    const float4 v = ((const float4*)(x + (size_t)s * FEAT))[lane];
    float* ap = agg + (size_t)d * FEAT + lane * 4;
    atomicAdd(ap + 0, v.x);
    atomicAdd(ap + 1, v.y);
    atomicAdd(ap + 2, v.z);
    atomicAdd(ap + 3, v.w);
    if (add_deg && lane == 0) atomicAdd(deg + d, 1.0f);
}

// -------------------------------------------------- fused SAGE layer (WMMA)
// out[N,M] = relu( (agg/deg) @ Wnb + b + xs @ Wself )
// One 16x16 tile per wave. A 16x4 f32 fragment: lane&15 = row, (lane>>4)*2 =
// K sub-offset (folded into base pointers), 2 VGPRs/lane. B 4x16 mirrors it
// with lane&15 = col. C/D: VGPR i holds row i (+8 for upper half-wave).
template <int K, int M>
__global__ void gs_sage_wmma_kernel(const float* __restrict__ agg,
                                    const float* __restrict__ deg,
                                    const float* __restrict__ xs,
                                    const float* __restrict__ Wnb,
                                    const float* __restrict__ Wself,
                                    const float* __restrict__ bias,
                                    float* __restrict__ out, int N) {
    const int lane   = threadIdx.x & 31;
    const int wave   = threadIdx.x >> 5;
    constexpr int tilesM = M / 16;
    const int total  = (N >> 4) * tilesM;
    const int t      = blockIdx.x * (blockDim.x >> 5) + wave;   // wave-uniform
    if (t >= total) return;                                      // whole wave exits

    const int tn   = t / tilesM;
    const int tm   = t - tn * tilesM;
    const int arow = tn * 16 + (lane & 15);
    const int bcol = tm * 16 + (lane & 15);
    const int koff = (lane >> 4) << 1;          // 0 for lanes 0-15, 2 for 16-31

    const float invd = 1.0f / fmaxf(deg[arow], 1.0f);
    // Fold per-lane koff into base pointers -> uniform loop bounds below.
    const float* __restrict__ rowA = agg + (size_t)arow * K + koff;
    const float* __restrict__ rowX = xs  + (size_t)arow * K + koff;
    const float* __restrict__ Bnb  = Wnb   + (size_t)koff * M + bcol;
    const float* __restrict__ Bsf  = Wself + (size_t)koff * M + bcol;

    v8f c = {};
    // Pass 1: mean-aggregated neighbors @ W_nb (mean formed on the fly)
#pragma unroll
    for (int kk = 0; kk < K; kk += 4) {
        v2f a = *(const v2f*)(rowA + kk);       // 8B aligned (koff even)
        a.x *= invd; a.y *= invd;
        v2f b; b.x = Bnb[(size_t)kk * M]; b.y = Bnb[(size_t)(kk + 1) * M];
        c = __builtin_amdgcn_wmma_f32_16x16x4_f32(false, a, false, b,
                                                  (short)0, c, false, false);
    }
    // Pass 2: self features @ W_self
#pragma unroll
    for (int kk = 0; kk < K; kk += 4) {
        v2f a = *(const v2f*)(rowX + kk);
        v2f b; b.x = Bsf[(size_t)kk * M]; b.y = Bsf[(size_t)(kk + 1) * M];
        c = __builtin_amdgcn_wmma_f32_16x16x4_f32(false, a, false, b,
                                                  (short)0, c, false, false);
    }

    const float bv    = bias[bcol];
    const int   rbase = tn * 16 + (lane >> 4) * 8;
#pragma unroll
    for (int i = 0; i < 8; ++i) {
        float v = fmaxf(c[i] + bv, 0.0f);        // fused bias + ReLU
        out[(size_t)(rbase + i) * M + bcol] = v;
    }
}

// --------------------------------------------------------- FC head (WMMA)
// logits[N,M] = h @ Wfc + bfc with M=40 (not a multiple of 16): 3 column
// tiles with per-lane column guards. Guards only touch loads/stores; the
// WMMA itself always runs with full EXEC (guard branches re-converge first).
template <int K, int M, int TILES_M>
__global__ void gs_fc_wmma_kernel(const float* __restrict__ h,
                                  const float* __restrict__ W,
                                  const float* __restrict__ bias,
                                  float* __restrict__ out, int N) {
    const int lane  = threadIdx.x & 31;
    const int wave  = threadIdx.x >> 5;
    const int total = (N >> 4) * TILES_M;
    const int t     = blockIdx.x * (blockDim.x >> 5) + wave;    // wave-uniform
    if (t >= total) return;

    const int tn    = t / TILES_M;
    const int tm    = t - tn * TILES_M;
    const int arow  = tn * 16 + (lane & 15);
    const int bcol  = tm * 16 + (lane & 15);
    const bool cok  = bcol < M;
    const int koff  = (lane >> 4) << 1;

    const float* __restrict__ rowH = h + (size_t)arow * K + koff;
    // Clamp the guarded column so OOB lanes still have a valid address; their
    // B values are forced to 0 so they contribute nothing to the product.
    const int bcc = cok ? bcol : (M - 1);
    const float* __restrict__ Bw = W + (size_t)koff * M + bcc;

    v8f c = {};
#pragma unroll
    for (int kk = 0; kk < K; kk += 4) {
        v2f a = *(const v2f*)(rowH + kk);
        v2f b;
        b.x = cok ? Bw[(size_t)kk * M]       : 0.0f;
        b.y = cok ? Bw[(size_t)(kk + 1) * M] : 0.0f;
        c = __builtin_amdgcn_wmma_f32_16x16x4_f32(false, a, false, b,
                                                  (short)0, c, false, false);
    }

    if (cok) {
        const float bv    = bias[bcol];
        const int   rbase = tn * 16 + (lane >> 4) * 8;
#pragma unroll
        for (int i = 0; i < 8; ++i)
            out[(size_t)(rbase + i) * M + bcol] = c[i] + bv;
    }
}

// ---------------------------------------------------------------------------
extern "C" void kernel_launch(void* const* d_in, const int* in_sizes, int n_in,
                              void* d_out, int out_size, void* d_ws, size_t ws_size,
                              hipStream_t stream) {
    const float* x       = (const float*)d_in[0];   // [40000,128]
    const int*   eidx    = (const int*)  d_in[1];   // [2,640000]
    const float* W1_nb   = (const float*)d_in[2];   // [128,128]
    const float* b1      = (const float*)d_in[3];   // [128]
    const float* W1_self = (const float*)d_in[4];   // [128,128]
    const float* W2_nb   = (const float*)d_in[5];   // [128,64]
    const float* b2      = (const float*)d_in[6];   // [64]
    const float* W2_self = (const float*)d_in[7];   // [128,64]
    const float* Wfc     = (const float*)d_in[8];   // [64,40]
    const float* bfc     = (const float*)d_in[9];   // [40]

    const int n_edges = in_sizes[1] / 2;
    const int* src = eidx;
    const int* dst = eidx + n_edges;

    // Workspace layout: agg[40000*128] | h1[40000*128] | deg[40000]  (~41.1MB)
    float* agg = (float*)d_ws;
    float* h1  = agg + (size_t)N_NODES * FEAT;
    float* deg = h1  + (size_t)N_NODES * FEAT;

    float* emb    = (float*)d_out;                        // [40000,64]
    float* logits = emb + (size_t)N_NODES * 64;           // [40000,40]

    const int agg_elems = N_NODES * FEAT;                 // 5,120,000

    // ---- Layer 1: zero accumulators, scatter x, fused SAGE GEMM -> h1
    gs_zero_kernel<<<(agg_elems + 255) / 256, 256, 0, stream>>>(agg, agg_elems);
    gs_zero_kernel<<<(N_NODES + 255) / 256, 256, 0, stream>>>(deg, N_NODES);
    {
        long long threads = (long long)n_edges * 32;
        gs_scatter_kernel<<<(unsigned)((threads + 255) / 256), 256, 0, stream>>>(
            x, src, dst, agg, deg, n_edges, /*add_deg=*/1);
    }
    // tiles = (40000/16) * (128/16) = 20000 -> 8 waves/block -> 2500 blocks
    gs_sage_wmma_kernel<128, 128><<<2500, 256, 0, stream>>>(
        agg, deg, x, W1_nb, W1_self, b1, h1, N_NODES);

    // ---- Layer 2: re-zero agg, scatter h1, fused SAGE GEMM -> embedding
    gs_zero_kernel<<<(agg_elems + 255) / 256, 256, 0, stream>>>(agg, agg_elems);
    {
        long long threads = (long long)n_edges * 32;
        gs_scatter_kernel<<<(unsigned)((threads + 255) / 256), 256, 0, stream>>>(
            h1, src, dst, agg, deg, n_edges, /*add_deg=*/0);
    }
    // tiles = 2500 * (64/16) = 10000 -> 1250 blocks
    gs_sage_wmma_kernel<128, 64><<<1250, 256, 0, stream>>>(
        agg, deg, h1, W2_nb, W2_self, b2, emb, N_NODES);

    // ---- FC head: logits = emb @ Wfc + bfc  (M=40, 3 guarded column tiles)
    {
        const int total  = (N_NODES / 16) * 3;            // 7500 tiles
        const int blocks = (total + 7) / 8;               // 8 waves per block
        gs_fc_wmma_kernel<64, 40, 3><<<blocks, 256, 0, stream>>>(
            emb, Wfc, bfc, logits, N_NODES);
    }
}